// TopoLossMSE2D_2740189134902
// MI455X (gfx1250) — compile-verified
//
#include <hip/hip_runtime.h>
#include <hip/hip_bf16.h>
#include <float.h>

// Problem constants (reference: pred/target (16,3,500,500) fp32, TOPO=100)
#define TOPO        100
#define IMG_H       500
#define IMG_W       500
#define PH          5
#define PW          5
#define NBC         48                 // 16*3 (batch*channel) slices
#define NPATCH      (NBC * PH * PW)    // 1200 patches
#define PATCH_ELEMS (TOPO * TOPO)      // 10000 floats per patch
#define THREADS     256
#define NWAVES      (THREADS / 32)

typedef unsigned int v4u __attribute__((ext_vector_type(4)));
typedef int          v4i __attribute__((ext_vector_type(4)));
typedef int          v8i __attribute__((ext_vector_type(8)));

struct __align__(16) SharedBlk {
    float tile[PATCH_ELEMS];   // 40000 B, filled by TDM (packed row-major patch)
    float wmaxv[NWAVES]; int wmaxi[NWAVES];
    float wminv[NWAVES]; int wmini[NWAVES];
};

__device__ __forceinline__ void upd_max(float v, int i, float& mv, int& mi) {
    if (v > mv || (v == mv && i < mi)) { mv = v; mi = i; }
}
__device__ __forceinline__ void upd_min(float v, int i, float& mv, int& mi) {
    if (v < mv || (v == mv && i < mi)) { mv = v; mi = i; }
}

__global__ void __launch_bounds__(THREADS)
topo_patch_kernel(const float* __restrict__ pred,
                  const float* __restrict__ target,
                  float* __restrict__ ws) {
    __shared__ SharedBlk sh;

    const int pid = blockIdx.x;                 // 0..1199
    const int bc  = pid / (PH * PW);            // (batch,channel) slice
    const int pp  = pid % (PH * PW);
    const int pr  = pp / PW;
    const int pc  = pp % PW;
    // element offset of patch origin inside the flat (B*C,H,W) tensor
    const long long base = (long long)bc * (IMG_H * IMG_W)
                         + (long long)pr * TOPO * IMG_W
                         + (long long)pc * TOPO;

    // --- Wave 0 issues the Tensor Data Mover load: 100x100 fp32 tile,
    //     row stride 500 elements, packed contiguously into LDS. ---
    if (threadIdx.x < 32) {
        unsigned long long ga = (unsigned long long)(const void*)(pred + base);
        // Flat LDS pointer low 32 bits == wave-relative LDS byte offset (ISA 10.2)
        unsigned ldsa = (unsigned)(unsigned long long)(void*)sh.tile;

        v4u g0;
        g0[0] = 1u;                                        // count=1 (valid D#)
        g0[1] = ldsa;                                      // lds_addr [63:32]
        g0[2] = (unsigned)(ga & 0xFFFFFFFFu);              // global_addr lo
        g0[3] = (unsigned)((ga >> 32) & 0x01FFFFFFu)       // global_addr[56:32]
              | 0x80000000u;                               // type=2 ("image") [127:126]

        v8i g1;
        g1[0] = (int)(2u << 16);                           // data_size=4B; wg_mask=0
        g1[1] = (int)((IMG_W & 0xFFFF) << 16);             // tensor_dim0 = 500  [79:48]
        g1[2] = (int)((IMG_W >> 16) | ((TOPO & 0xFFFF) << 16)); // tensor_dim1 = 100 [111:80]
        g1[3] = (int)((TOPO >> 16) | (TOPO << 16));        // tile_dim0 = 100 [127:112]
        g1[4] = (int)TOPO;                                 // tile_dim1 = 100, tile_dim2 = 0
        g1[5] = (int)IMG_W;                                // tensor_dim0_stride = 500
        g1[6] = 0;
        g1[7] = 0;                                         // tensor_dim1_stride unused (2D)

        v4i z4; z4[0] = 0; z4[1] = 0; z4[2] = 0; z4[3] = 0;  // groups 2/3 unused (<=2D)
        v8i z8; z8[0] = 0; z8[1] = 0; z8[2] = 0; z8[3] = 0;
                z8[4] = 0; z8[5] = 0; z8[6] = 0; z8[7] = 0;

        // clang-23 / amdgpu-toolchain 6-arg form:
        // (v4u g0, v8i g1, v4i g2, v4i g3, v8i extra, i32 cpol)
        __builtin_amdgcn_tensor_load_to_lds(g0, g1, z4, z4, z8, 0);
        __builtin_amdgcn_s_wait_tensorcnt(0);              // s_wait_tensorcnt 0
    }
    __syncthreads();

    // --- Scan LDS tile: per-thread argmax/argmin over float4 chunks ---
    float mxv = -FLT_MAX; int mxi = 0x7FFFFFFF;
    float mnv =  FLT_MAX; int mni = 0x7FFFFFFF;
    const float4* t4 = reinterpret_cast<const float4*>(sh.tile);
    for (int q = threadIdx.x; q < PATCH_ELEMS / 4; q += THREADS) {
        float4 v = t4[q];                                  // ds_load_b128
        const int k = q * 4;
        upd_max(v.x, k + 0, mxv, mxi); upd_min(v.x, k + 0, mnv, mni);
        upd_max(v.y, k + 1, mxv, mxi); upd_min(v.y, k + 1, mnv, mni);
        upd_max(v.z, k + 2, mxv, mxi); upd_min(v.z, k + 2, mnv, mni);
        upd_max(v.w, k + 3, mxv, mxi); upd_min(v.w, k + 3, mnv, mni);
    }

    // --- wave32 butterfly reduction (index tie-break keeps it deterministic) ---
    for (int off = 16; off > 0; off >>= 1) {
        float ov = __shfl_xor(mxv, off, 32);
        int   oi = __shfl_xor(mxi, off, 32);
        upd_max(ov, oi, mxv, mxi);
        float pv = __shfl_xor(mnv, off, 32);
        int   pi = __shfl_xor(mni, off, 32);
        upd_min(pv, pi, mnv, mni);
    }

    const int wid  = threadIdx.x >> 5;
    const int lane = threadIdx.x & 31;
    if (lane == 0) {
        sh.wmaxv[wid] = mxv; sh.wmaxi[wid] = mxi;
        sh.wminv[wid] = mnv; sh.wmini[wid] = mni;
    }
    __syncthreads();

    if (threadIdx.x == 0) {
        float bmv = sh.wmaxv[0]; int bmi = sh.wmaxi[0];
        float dmv = sh.wminv[0]; int dmi = sh.wmini[0];
        #pragma unroll
        for (int w = 1; w < NWAVES; ++w) {
            upd_max(sh.wmaxv[w], sh.wmaxi[w], bmv, bmi);
            upd_min(sh.wminv[w], sh.wmini[w], dmv, dmi);
        }
        // Map flattened patch indices back to the global tensor for target fetch
        const int br = bmi / TOPO, bcol = bmi % TOPO;
        const int dr = dmi / TOPO, dcol = dmi % TOPO;
        const float tb = target[base + (long long)br * IMG_W + bcol];
        const float td = target[base + (long long)dr * IMG_W + dcol];
        const float eb = bmv - tb;
        const float ed = dmv - td;
        ws[pid] = eb * eb + ed * ed;
    }
}

__global__ void __launch_bounds__(THREADS)
topo_reduce_kernel(const float* __restrict__ ws, float* __restrict__ out) {
    __shared__ float s[NWAVES];
    float acc = 0.f;
    for (int i = threadIdx.x; i < NPATCH; i += THREADS) acc += ws[i];
    for (int off = 16; off > 0; off >>= 1) acc += __shfl_xor(acc, off, 32);
    const int wid  = threadIdx.x >> 5;
    const int lane = threadIdx.x & 31;
    if (lane == 0) s[wid] = acc;
    __syncthreads();
    if (threadIdx.x == 0) {
        float t = 0.f;
        #pragma unroll
        for (int w = 0; w < NWAVES; ++w) t += s[w];
        out[0] = t;
    }
}

extern "C" void kernel_launch(void* const* d_in, const int* in_sizes, int n_in,
                              void* d_out, int out_size, void* d_ws, size_t ws_size,
                              hipStream_t stream) {
    const float* pred   = (const float*)d_in[0];
    const float* target = (const float*)d_in[1];
    float*       ws     = (float*)d_ws;          // 1200 floats of scratch
    float*       out    = (float*)d_out;

    topo_patch_kernel<<<NPATCH, THREADS, 0, stream>>>(pred, target, ws);
    topo_reduce_kernel<<<1, THREADS, 0, stream>>>(ws, out);
}